// EnhancedRGCN_50483045597788
// MI455X (gfx1250) — compile-verified
//
#include <hip/hip_runtime.h>
#include <math.h>

// ---------------------------------------------------------------------------
// 3-layer GATConv pipeline for MI455X (gfx1250).
//  - Dense 32->32 projections use V_WMMA_F32_16X16X4_F32 (fp32 matrix pipe).
//  - Edge softmax/aggregate: 3 streaming passes per layer; node state is
//    L2-resident (192 MB L2 >> 30 MB working set); atomics resolve at L2.
// ---------------------------------------------------------------------------

typedef __attribute__((ext_vector_type(2))) float v2f;
typedef __attribute__((ext_vector_type(8))) float v8f;

// Order-preserving float<->uint encoding so segment-max can use atomicMax(u32)
// (lowers to GLOBAL_ATOMIC_MAX_U32). enc(x) > 0 for every finite x, so a
// 0-initialized buffer acts as -inf identity.
__device__ __forceinline__ unsigned fenc(float f) {
  unsigned u = __float_as_uint(f);
  return (u & 0x80000000u) ? ~u : (u | 0x80000000u);
}
__device__ __forceinline__ float fdec(unsigned u) {
  return (u & 0x80000000u) ? __uint_as_float(u & 0x7fffffffu)
                           : __uint_as_float(~u);
}

__device__ __forceinline__ float lrelu(float x, float slope) {
  return x > 0.0f ? x : slope * x;
}

// ---------------------------------------------------------------------------
// h[N,32] = X[N,32] @ W[32,32]^T  via V_WMMA_F32_16X16X4_F32.
// One wave -> one 16-row stripe -> two 16x16 output tiles, K=32 in steps of 4.
// ---------------------------------------------------------------------------
__global__ void __launch_bounds__(256)
gat_gemm32_wmma(const float* __restrict__ X, const float* __restrict__ W,
                float* __restrict__ H, int N) {
  int wave = (int)((blockIdx.x * blockDim.x + threadIdx.x) >> 5);
  int lane = (int)(threadIdx.x & 31);
  int m0 = wave << 4;
  if (m0 >= N) return;  // wave-granular: EXEC stays all-1s for WMMA

  int row = m0 + (lane & 15);
  int kh  = (lane >> 4) << 1;   // ISA 16x4 A layout: hi half-wave holds K+2,K+3
  int col = lane & 15;

  v8f acc0 = {0.f,0.f,0.f,0.f,0.f,0.f,0.f,0.f};
  v8f acc1 = {0.f,0.f,0.f,0.f,0.f,0.f,0.f,0.f};
  const float* xr = X + (size_t)row * 32;

#pragma unroll
  for (int kk = 0; kk < 32; kk += 4) {
    v2f a, b0, b1;
    a.x  = xr[kk + kh];
    a.y  = xr[kk + kh + 1];
    // B[k][n] = W[n][k]  (h = x @ W^T), cols 0..15 and 16..31
    b0.x = W[(size_t)col * 32 + kk + kh];
    b0.y = W[(size_t)col * 32 + kk + kh + 1];
    b1.x = W[(size_t)(col + 16) * 32 + kk + kh];
    b1.y = W[(size_t)(col + 16) * 32 + kk + kh + 1];
    acc0 = __builtin_amdgcn_wmma_f32_16x16x4_f32(false, a, false, b0,
                                                 (short)0, acc0, false, false);
    acc1 = __builtin_amdgcn_wmma_f32_16x16x4_f32(false, a, false, b1,
                                                 (short)0, acc1, false, false);
  }

  // C/D layout: VGPR r, lanes 0-15 -> M=r, lanes 16-31 -> M=r+8
  int orow = m0 + ((lane >> 4) << 3);
#pragma unroll
  for (int r = 0; r < 8; ++r) {
    H[(size_t)(orow + r) * 32 + col]      = acc0[r];
    H[(size_t)(orow + r) * 32 + col + 16] = acc1[r];
  }
}

// a_s[n,h] = sum_c h[n,h*ch+c]*att_s[h,c]; same for a_d. Stride-2 storage.
__global__ void attn_scores(const float* __restrict__ H,
                            const float* __restrict__ atts,
                            const float* __restrict__ attd,
                            float* __restrict__ as_, float* __restrict__ ad_,
                            int N, int heads) {
  int n = (int)(blockIdx.x * blockDim.x + threadIdx.x);
  if (n >= N) return;
  int ch = 32 / heads;
  for (int h = 0; h < heads; ++h) {
    float ss = 0.f, sd = 0.f;
    for (int c = 0; c < ch; ++c) {
      float v = H[(size_t)n * 32 + h * ch + c];
      ss += v * atts[h * ch + c];
      sd += v * attd[h * ch + c];
    }
    as_[(size_t)n * 2 + h] = ss;
    ad_[(size_t)n * 2 + h] = sd;
  }
}

__global__ void init_state(float* __restrict__ accum, unsigned* __restrict__ m,
                           float* __restrict__ denom, int N) {
  int i = (int)(blockIdx.x * blockDim.x + threadIdx.x);
  if (i < N * 32) accum[i] = 0.f;
  if (i < N * 2) { m[i] = 0u; denom[i] = 0.f; }
}

// Pass A: segment max of leaky-relu logits over dst.
__global__ void edge_max(const int* __restrict__ src, const int* __restrict__ dst,
                         const float* __restrict__ as_, const float* __restrict__ ad_,
                         unsigned* __restrict__ m, long long E, int heads,
                         float slope) {
  long long e = blockIdx.x * (long long)blockDim.x + threadIdx.x;
  if (e >= E) return;
  int s = src[e], d = dst[e];
  for (int h = 0; h < heads; ++h) {
    float l = lrelu(as_[(size_t)s * 2 + h] + ad_[(size_t)d * 2 + h], slope);
    atomicMax(&m[(size_t)d * 2 + h], fenc(l));
  }
}

// Pass B: denom[dst] += exp(logit - max)
__global__ void edge_expsum(const int* __restrict__ src, const int* __restrict__ dst,
                            const float* __restrict__ as_, const float* __restrict__ ad_,
                            const unsigned* __restrict__ m, float* __restrict__ denom,
                            long long E, int heads, float slope) {
  long long e = blockIdx.x * (long long)blockDim.x + threadIdx.x;
  if (e >= E) return;
  int s = src[e], d = dst[e];
  for (int h = 0; h < heads; ++h) {
    float l = lrelu(as_[(size_t)s * 2 + h] + ad_[(size_t)d * 2 + h], slope);
    float ex = __expf(l - fdec(m[(size_t)d * 2 + h]));
    atomicAdd(&denom[(size_t)d * 2 + h], ex);
  }
}

// Pass C: accum[dst,:] += alpha * h[src,:]. 4 threads/edge, 8 channels each,
// float4 gathers (B128 loads) + 8 fp32 atomics (resolve at L2).
__global__ void edge_aggregate(const int* __restrict__ src, const int* __restrict__ dst,
                               const float* __restrict__ as_, const float* __restrict__ ad_,
                               const unsigned* __restrict__ m,
                               const float* __restrict__ denom,
                               const float* __restrict__ H,
                               float* __restrict__ accum, long long E, int heads,
                               float slope) {
  long long t = blockIdx.x * (long long)blockDim.x + threadIdx.x;
  long long e = t >> 2;
  if (e >= E) return;
  int part = (int)(t & 3);
  int s = src[e], d = dst[e];
  int ch = 32 / heads;
  int h = (part * 8) / ch;  // 8 <= ch, so one head per thread

  float l = lrelu(as_[(size_t)s * 2 + h] + ad_[(size_t)d * 2 + h], slope);
  float ex = __expf(l - fdec(m[(size_t)d * 2 + h]));
  float alpha = ex / (denom[(size_t)d * 2 + h] + 1e-16f);

  const float4* hs = (const float4*)(H + (size_t)s * 32 + part * 8);
  float4 v0 = hs[0], v1 = hs[1];
  float* out = accum + (size_t)d * 32 + part * 8;
  atomicAdd(out + 0, alpha * v0.x);
  atomicAdd(out + 1, alpha * v0.y);
  atomicAdd(out + 2, alpha * v0.z);
  atomicAdd(out + 3, alpha * v0.w);
  atomicAdd(out + 4, alpha * v1.x);
  atomicAdd(out + 5, alpha * v1.y);
  atomicAdd(out + 6, alpha * v1.z);
  atomicAdd(out + 7, alpha * v1.w);
}

// mode 0: layer1 (+bias, tanh-gate scale, *1.05, elu, clip)
// mode 1: +bias, elu, clip
// mode 2: +bias only (final output)
__global__ void epilogue(const float* __restrict__ accum,
                         const float* __restrict__ bias,
                         const float* __restrict__ ea, float* __restrict__ out,
                         int N, int mode) {
  int i = (int)(blockIdx.x * blockDim.x + threadIdx.x);
  if (i >= N * 32) return;
  int c = i & 31;
  float v = accum[i] + bias[c];
  if (mode == 0) {
    float sc = tanhf(ea[0]);
    if (sc < 0.1f) sc = 1.0f;   // 'continue' branch skips scaling
    v = v * sc * 1.05f;          // h + 0.05*h.detach()
  }
  if (mode <= 1) {
    v = v > 0.f ? v : expm1f(v); // elu
    v = fminf(3.0f, fmaxf(-3.0f, v));
  }
  out[i] = v;
}

extern "C" void kernel_launch(void* const* d_in, const int* in_sizes, int n_in,
                              void* d_out, int out_size, void* d_ws, size_t ws_size,
                              hipStream_t stream) {
  (void)n_in; (void)out_size; (void)ws_size;

  const float* x    = (const float*)d_in[0];
  const int*   ei   = (const int*)d_in[1];
  const float* W1   = (const float*)d_in[2];
  const float* as1  = (const float*)d_in[3];
  const float* ad1  = (const float*)d_in[4];
  const float* b1   = (const float*)d_in[5];
  const float* ea1  = (const float*)d_in[6];
  const float* W2   = (const float*)d_in[7];
  const float* as2  = (const float*)d_in[8];
  const float* ad2  = (const float*)d_in[9];
  const float* b2   = (const float*)d_in[10];
  const float* W3   = (const float*)d_in[11];
  const float* as3  = (const float*)d_in[12];
  const float* ad3  = (const float*)d_in[13];
  const float* b3   = (const float*)d_in[14];

  const int N = in_sizes[0] / 32;
  const long long E = in_sizes[1] / 2;
  const int* srcp = ei;
  const int* dstp = ei + E;

  // workspace partition (floats)
  float*    hA    = (float*)d_ws;                 // N*32  (post-GEMM h)
  float*    hB    = hA + (size_t)N * 32;          // N*32  (layer output / next input)
  float*    accum = hB + (size_t)N * 32;          // N*32
  float*    as_   = accum + (size_t)N * 32;       // N*2
  float*    ad_   = as_ + (size_t)N * 2;          // N*2
  unsigned* m     = (unsigned*)(ad_ + (size_t)N * 2);  // N*2
  float*    denom = (float*)(m + (size_t)N * 2);       // N*2

  const dim3 blk(256);
  const int gridGemm  = (int)((((N + 15) / 16) + 7) / 8);  // 8 waves / block
  const int gridNode  = (N + 255) / 256;
  const int gridElem  = (N * 32 + 255) / 256;
  const int gridE     = (int)((E + 255) / 256);
  const int gridE4    = (int)((4 * E + 255) / 256);

  auto run_layer = [&](const float* xin, const float* Wt, const float* atts,
                       const float* attd, const float* bias, int heads,
                       float slope, int mode, float* outbuf) {
    gat_gemm32_wmma<<<gridGemm, blk, 0, stream>>>(xin, Wt, hA, N);
    attn_scores<<<gridNode, blk, 0, stream>>>(hA, atts, attd, as_, ad_, N, heads);
    init_state<<<gridElem, blk, 0, stream>>>(accum, m, denom, N);
    edge_max<<<gridE, blk, 0, stream>>>(srcp, dstp, as_, ad_, m, E, heads, slope);
    edge_expsum<<<gridE, blk, 0, stream>>>(srcp, dstp, as_, ad_, m, denom, E, heads, slope);
    edge_aggregate<<<gridE4, blk, 0, stream>>>(srcp, dstp, as_, ad_, m, denom,
                                               hA, accum, E, heads, slope);
    epilogue<<<gridElem, blk, 0, stream>>>(accum, bias, ea1, outbuf, N, mode);
  };

  // layer 1: heads=2, ch=16, neg_slope=0.01, gated-scale + elu + clip
  run_layer(x,  W1, as1, ad1, b1, 2, 0.01f, 0, hB);
  // layer 2: heads=2, ch=16, neg_slope=0.2, elu + clip
  run_layer(hB, W2, as2, ad2, b2, 2, 0.20f, 1, hB);
  // layer 3: heads=1, ch=32, neg_slope=0.2, bias only
  run_layer(hB, W3, as3, ad3, b3, 1, 0.20f, 2, (float*)d_out);
}